// GMM_13889924235867
// MI455X (gfx1250) — compile-verified
//
#include <hip/hip_runtime.h>
#include <hip/hip_bf16.h>

// GMM forward for MI455X (gfx1250, wave32).
// Main compute: per-cluster maha via V_WMMA_F32_16X16X4_F32 (fp32 WMMA).
// Each wave processes 32 points (2 WMMA N-tiles) so the A-matrix (M_k) LDS
// loads are amortized over two B tiles: 6 DS loads per 8 WMMAs.

typedef __attribute__((ext_vector_type(2))) float v2f;
typedef __attribute__((ext_vector_type(8))) float v8f;

#define LOG2PI_F 1.8378770664093454f

// ---------------- zero output ----------------
__global__ void gmm_zero_kernel(float* __restrict__ out, int n) {
    int i = blockIdx.x * blockDim.x + threadIdx.x;
    if (i < n) out[i] = 0.0f;
}

// ---------------- per-cluster prep: cov, Cholesky, L^{-1}, constants -------
// grid = K blocks, 64 threads per block (thread = row / column index).
__global__ __launch_bounds__(64) void gmm_prep_kernel(
    const float* __restrict__ weights,      // [K]
    const float* __restrict__ covs,         // [K,64,64]
    float* __restrict__ Mmat,               // [K,64,64]  (L^{-1}, dense)
    float* __restrict__ constk)             // [K]
{
    __shared__ float a[64 * 65];     // cov -> Cholesky L (in place, lower)
    __shared__ float Minv[64 * 65];  // L^{-1}

    const int tid = threadIdx.x;
    const int k = blockIdx.x;
    const float* C = covs + (size_t)k * 4096;

    // cov = C C^T + I   (row tid)
    for (int j = 0; j < 64; ++j) {
        float s = (tid == j) ? 1.0f : 0.0f;
        for (int l = 0; l < 64; ++l) s += C[tid * 64 + l] * C[j * 64 + l];
        a[tid * 65 + j] = s;
    }
    __syncthreads();

    // Right-looking Cholesky (lower triangular)
    for (int j = 0; j < 64; ++j) {
        if (tid == 0) a[j * 65 + j] = sqrtf(a[j * 65 + j]);
        __syncthreads();
        if (tid > j) a[tid * 65 + j] /= a[j * 65 + j];
        __syncthreads();
        for (int l = j + 1; l < 64; ++l)
            if (tid >= l) a[tid * 65 + l] -= a[tid * 65 + j] * a[l * 65 + j];
        __syncthreads();
    }

    // Triangular inverse: thread c computes column c of M = L^{-1}
    {
        const int c = tid;
        for (int i = 0; i < 64; ++i) Minv[i * 65 + c] = 0.0f;
        Minv[c * 65 + c] = 1.0f / a[c * 65 + c];
        for (int i = c + 1; i < 64; ++i) {
            float s = 0.0f;
            for (int l = c; l < i; ++l) s += a[i * 65 + l] * Minv[l * 65 + c];
            Minv[i * 65 + c] = -s / a[i * 65 + i];
        }
    }
    __syncthreads();

    // Write dense M (row tid)
    for (int j = 0; j < 64; ++j)
        Mmat[(size_t)k * 4096 + tid * 64 + j] = Minv[tid * 65 + j];

    if (tid == 0) {
        float logdet = 0.0f;
        for (int j = 0; j < 64; ++j) logdet += logf(a[j * 65 + j]);
        logdet *= 2.0f;
        constk[k] = -0.5f * (64.0f * LOG2PI_F + logdet) + logf(weights[k]);
    }
}

// ---------------- main: maha via fp32 WMMA, argmax, xo scatter, loss -------
// Block = 128 threads = 4 waves, handling 128 consecutive points (one batch
// entry since 128 | 16384). Wave w handles points [32w, 32w+32): two 16-point
// WMMA N-tiles sharing the A-matrix loads. For each cluster it computes
// Y = M_k (x - mu_k)^T as 4 row-blocks x 2 point-tiles of 16x16 WMMA output,
// then maha = column sums of Y^2.
__global__ __launch_bounds__(128) void gmm_maha_kernel(
    const float* __restrict__ x,        // [N,64]
    const float* __restrict__ Mmat,     // [K,64,64]
    const float* __restrict__ means,    // [K,64]
    const float* __restrict__ constk,   // [K]
    float* __restrict__ out)            // [0..8191] xo [B,K,64], [8192] loss
{
    __shared__ float xs[128 * 65];   // x tile, padded (conflict-free)
    __shared__ float Ms[64 * 65];    // M_k, padded
    __shared__ float mus[64];
    __shared__ float xo_s[16 * 64];  // per-block xo partial [K,64]
    __shared__ float ls[4];          // per-wave loss partials

    const int tid  = threadIdx.x;    // 0..127
    const int wave = tid >> 5;       // 0..3
    const int lane = tid & 31;
    const int half = lane >> 4;      // 0: lanes 0-15, 1: lanes 16-31
    const int lp   = lane & 15;
    const long long p0 = (long long)blockIdx.x * 128;

    // Stage x tile (coalesced global read, padded LDS rows)
    for (int i = tid; i < 128 * 64; i += 128) {
        int r = i >> 6, c = i & 63;
        xs[r * 65 + c] = x[p0 * 64 + i];
    }
    for (int i = tid; i < 16 * 64; i += 128) xo_s[i] = 0.0f;

    float best0 = -3.402823e38f, best1 = -3.402823e38f;
    int   bidx0 = 0, bidx1 = 0;
    float lsum = 0.0f;

    const int prow0 = 32 * wave + lp;        // point row of tile 0
    const int prow1 = prow0 + 16;            // point row of tile 1

    for (int k = 0; k < 16; ++k) {
        __syncthreads();  // protect Ms/mus reuse + (k==0) xs/xo_s init
        for (int i = tid; i < 64 * 64; i += 128) {
            int r = i >> 6, c = i & 63;
            Ms[r * 65 + c] = Mmat[(size_t)k * 4096 + i];
        }
        if (tid < 64) mus[tid] = means[k * 64 + tid];
        __syncthreads();

        v8f c0 = {}, c1 = {}, c2 = {}, c3 = {};   // tile 0, row-blocks 0..3
        v8f c4 = {}, c5 = {}, c6 = {}, c7 = {};   // tile 1, row-blocks 0..3
        #pragma unroll
        for (int d4 = 0; d4 < 16; ++d4) {
            // A layout (16x4 f32): lanes 0-15 hold K={0,1}, lanes 16-31 K={2,3}
            // B layout (4x16 f32): mirrored; all contiguous float2 LDS loads.
            const int cb = 4 * d4 + 2 * half;
            const float m0 = mus[cb], m1 = mus[cb + 1];
            v2f b0, b1;
            b0.x = xs[prow0 * 65 + cb]     - m0;
            b0.y = xs[prow0 * 65 + cb + 1] - m1;
            b1.x = xs[prow1 * 65 + cb]     - m0;
            b1.y = xs[prow1 * 65 + cb + 1] - m1;
            v2f a0, a1, a2, a3;
            a0.x = Ms[(lp     ) * 65 + cb]; a0.y = Ms[(lp     ) * 65 + cb + 1];
            a1.x = Ms[(lp + 16) * 65 + cb]; a1.y = Ms[(lp + 16) * 65 + cb + 1];
            a2.x = Ms[(lp + 32) * 65 + cb]; a2.y = Ms[(lp + 32) * 65 + cb + 1];
            a3.x = Ms[(lp + 48) * 65 + cb]; a3.y = Ms[(lp + 48) * 65 + cb + 1];
            c0 = __builtin_amdgcn_wmma_f32_16x16x4_f32(false, a0, false, b0, (short)0, c0, false, false);
            c1 = __builtin_amdgcn_wmma_f32_16x16x4_f32(false, a1, false, b0, (short)0, c1, false, false);
            c2 = __builtin_amdgcn_wmma_f32_16x16x4_f32(false, a2, false, b0, (short)0, c2, false, false);
            c3 = __builtin_amdgcn_wmma_f32_16x16x4_f32(false, a3, false, b0, (short)0, c3, false, false);
            c4 = __builtin_amdgcn_wmma_f32_16x16x4_f32(false, a0, false, b1, (short)0, c4, false, false);
            c5 = __builtin_amdgcn_wmma_f32_16x16x4_f32(false, a1, false, b1, (short)0, c5, false, false);
            c6 = __builtin_amdgcn_wmma_f32_16x16x4_f32(false, a2, false, b1, (short)0, c6, false, false);
            c7 = __builtin_amdgcn_wmma_f32_16x16x4_f32(false, a3, false, b1, (short)0, c7, false, false);
        }

        // Per-lane sum of squares over the 32 Y rows this lane holds, per tile
        float ss0 = 0.0f, ss1 = 0.0f;
        #pragma unroll
        for (int j = 0; j < 8; ++j) {
            ss0 += c0[j]*c0[j] + c1[j]*c1[j] + c2[j]*c2[j] + c3[j]*c3[j];
            ss1 += c4[j]*c4[j] + c5[j]*c5[j] + c6[j]*c6[j] + c7[j]*c7[j];
        }
        // Combine the two half-waves: lane p and lane p+16 hold the same point
        ss0 += __shfl_xor(ss0, 16, 32);
        ss1 += __shfl_xor(ss1, 16, 32);

        const float ck = constk[k];
        float wlp0 = -0.5f * ss0 + ck;
        float wlp1 = -0.5f * ss1 + ck;
        if (half == 0) {                    // lanes 0-15 own points
            lsum += wlp0 + wlp1;
            if (wlp0 > best0) { best0 = wlp0; bidx0 = k; }  // first max on ties
            if (wlp1 > best1) { best1 = wlp1; bidx1 = k; }
        }
    }

    // Scatter x rows into per-block xo partials keyed by argmax cluster
    for (int pt = 0; pt < 16; ++pt) {
        int i0 = __shfl(bidx0, pt, 32);     // broadcast from lane pt (<16)
        int i1 = __shfl(bidx1, pt, 32);
        int r0 = 32 * wave + pt;
        int r1 = r0 + 16;
        atomicAdd(&xo_s[i0 * 64 + 2 * lane],     xs[r0 * 65 + 2 * lane]);
        atomicAdd(&xo_s[i0 * 64 + 2 * lane + 1], xs[r0 * 65 + 2 * lane + 1]);
        atomicAdd(&xo_s[i1 * 64 + 2 * lane],     xs[r1 * 65 + 2 * lane]);
        atomicAdd(&xo_s[i1 * 64 + 2 * lane + 1], xs[r1 * 65 + 2 * lane + 1]);
    }

    // Loss: full-wave reduction (lanes 16-31 contribute 0)
    lsum += __shfl_xor(lsum, 16, 32);
    lsum += __shfl_xor(lsum,  8, 32);
    lsum += __shfl_xor(lsum,  4, 32);
    lsum += __shfl_xor(lsum,  2, 32);
    lsum += __shfl_xor(lsum,  1, 32);
    if (lane == 0) ls[wave] = lsum;
    __syncthreads();

    // Flush xo partial (scaled by 1/S) and loss partial
    const int b = (int)(p0 >> 14);          // 16384 points per batch entry
    for (int i = tid; i < 1024; i += 128)
        atomicAdd(&out[b * 1024 + i], xo_s[i] * (1.0f / 16384.0f));
    if (tid == 0) {
        float t = 0.0f;
        for (int w = 0; w < 4; ++w) t += ls[w];
        atomicAdd(&out[8192], t * (-1.0f / 2097152.0f));  // -mean over B*S*K
    }
}

extern "C" void kernel_launch(void* const* d_in, const int* in_sizes, int n_in,
                              void* d_out, int out_size, void* d_ws, size_t ws_size,
                              hipStream_t stream) {
    const float* x       = (const float*)d_in[0];   // [8,16384,64]
    const float* weights = (const float*)d_in[1];   // [16]
    const float* means   = (const float*)d_in[2];   // [16,64]
    const float* covs    = (const float*)d_in[3];   // [16,64,64]
    float* out = (float*)d_out;                      // 8192 xo + 1 loss

    float* Mmat   = (float*)d_ws;                    // 16*4096 floats
    float* constk = Mmat + 16 * 4096;                // 16 floats

    gmm_zero_kernel<<<(8193 + 255) / 256, 256, 0, stream>>>(out, 8193);
    gmm_prep_kernel<<<16, 64, 0, stream>>>(weights, covs, Mmat, constk);
    gmm_maha_kernel<<<1024, 128, 0, stream>>>(x, Mmat, means, constk, out);
}